// CRFRNN_16346645529031
// MI455X (gfx1250) — compile-verified
//
#include <hip/hip_runtime.h>

// ---------------------------------------------------------------------------
// Fused CRF-RNN mean-field stencil for MI455X (gfx1250).
//
//   - One kernel, one pass: Gaussian weights computed once into registers,
//     5 message-passing iterations fused via overlapped (trapezoidal) tiling.
//   - EXT=512 extended positions/block (16 wave32) -> TILE=462 outputs
//     (halo = N_ITER*HALF = 25 per side, 90% thread efficiency).
//   - Double-buffered q exchange in LDS: ONE barrier per iteration.
//   - All input staging via the CDNA5 Tensor Data Mover:
//     tensor_load_to_lds + s_wait_tensorcnt (verified lowering last round);
//     plain-VMEM fallback kept for toolchains without the builtin.
// ---------------------------------------------------------------------------

constexpr int RAD    = 5;                      // HALF in the reference (K=11)
constexpr int NITER  = 5;
constexpr int EXT    = 512;                    // extended positions per block
constexpr int TILE   = EXT - 2 * NITER * RAD;  // 462 valid outputs per block
constexpr int FEATN  = EXT + 2 * RAD;          // 522 feature positions needed
constexpr int FEATF  = FEATN * 3;              // 1566 floats (interleaved xyz)
constexpr int QLEN   = EXT + 2 * RAD;          // 522 (q + halo pads)

typedef unsigned int u32;
typedef u32 u32x4 __attribute__((ext_vector_type(4)));
typedef int  i32x4 __attribute__((ext_vector_type(4)));
typedef int  i32x8 __attribute__((ext_vector_type(8)));

#if __has_builtin(__builtin_amdgcn_tensor_load_to_lds)
#define HAVE_TDM 1
#endif
#if __has_builtin(__builtin_amdgcn_s_wait_tensorcnt)
#define HAVE_WTC 1
#endif

#ifdef HAVE_TDM
// One TDM descriptor: copy `n` contiguous f32 from global memory into LDS.
// D# layout per CDNA5 ISA ch.8 (CDNA5 ISA):
//   group0: [1:0]=count=1, [63:32]=lds_addr, [120:64]=global_addr, [127:126]=2
//   group1: data_size=4B, tensor_dim0 = tile_dim0 = n, 1-D (tile_dim1/2 = 0)
__device__ __forceinline__ void tdm_load_1d_f32(float* lds_dst, const float* gsrc, u32 n) {
  // Low 32 bits of a flat pointer into LDS == wave-relative LDS byte address.
  u32 lds_off = (u32)(unsigned long long)(void*)lds_dst;
  unsigned long long ga = (unsigned long long)(const void*)gsrc;

  u32x4 g0;
  g0.x = 0x1u;                                           // count=1, user descriptor
  g0.y = lds_off;                                        // lds_addr (bytes)
  g0.z = (u32)(ga & 0xFFFFFFFFull);                      // global_addr[31:0]
  g0.w = (u32)((ga >> 32) & 0x01FFFFFFull) | (2u << 30); // addr[56:32] | type=2

  i32x8 g1;
  g1[0] = (int)(2u << 16);                               // data_size = 2 (4 bytes)
  g1[1] = (int)((n & 0xFFFFu) << 16);                    // tensor_dim0[15:0] @63:48
  g1[2] = (int)(((n >> 16) & 0xFFFFu) | (1u << 16));     // tensor_dim0[31:16]; tensor_dim1=1
  g1[3] = (int)((n & 0xFFFFu) << 16);                    // tensor_dim1 hi=0; tile_dim0=n
  g1[4] = 0;                                             // tile_dim1=0, tile_dim2=0 (1-D)
  g1[5] = (int)n;                                        // tensor_dim0_stride[31:0]
  g1[6] = 0;
  g1[7] = 0;

  i32x4 gz = {0, 0, 0, 0};
#if __clang_major__ >= 23
  i32x8 gz8 = {0, 0, 0, 0, 0, 0, 0, 0};
  __builtin_amdgcn_tensor_load_to_lds(g0, g1, gz, gz, gz8, 0);
#else
  __builtin_amdgcn_tensor_load_to_lds(g0, g1, gz, gz, 0);
#endif
}

__device__ __forceinline__ void tdm_wait0() {
#ifdef HAVE_WTC
  __builtin_amdgcn_s_wait_tensorcnt(0);
#else
  asm volatile("s_wait_tensorcnt 0x0" ::: "memory");
#endif
}
#endif // HAVE_TDM

__global__ __launch_bounds__(EXT)
void crfrnn_fused_kernel(const float* __restrict__ logits,
                         const float* __restrict__ p,
                         float* __restrict__ out, int N) {
  __shared__ float s_feat[FEATF];      // interleaved (pos, xyz)
  __shared__ float s_unary[EXT];
  __shared__ float s_q[2][QLEN];       // double-buffered; q at ext pos t -> [t+RAD]

  const int t     = threadIdx.x;
  const int b     = blockIdx.y;
  const int base  = (int)blockIdx.x * TILE - NITER * RAD; // ext pos t <-> g = base + t
  const int baseF = base - RAD;                           // first feat position

  // In-bounds sub-ranges of the LDS tiles (slots; the rest is zero-filled).
  const int uLo = (base < 0) ? -base : 0;
  int uHi = N - base;       if (uHi > EXT)   uHi = EXT;   if (uHi < uLo) uHi = uLo;
  const int fLo = (baseF < 0) ? -baseF : 0;
  int fHi = N - baseF;      if (fHi > FEATN) fHi = FEATN; if (fHi < fLo) fHi = fLo;

#ifdef HAVE_TDM
  // Zero only slots NOT covered by the DMA (disjoint addresses -> no race).
  for (int i = t; i < FEATF; i += EXT)
    if (i < fLo * 3 || i >= fHi * 3) s_feat[i] = 0.0f;
  if (t < uLo || t >= uHi) s_unary[t] = 0.0f;
  if (t < RAD) {
    s_q[0][t] = 0.0f; s_q[0][EXT + RAD + t] = 0.0f;   // halo pads, both buffers
    s_q[1][t] = 0.0f; s_q[1][EXT + RAD + t] = 0.0f;
  }

  if (t < 32) {  // wave 0 only (wave32): one DMA pair per workgroup
    if (fHi > fLo)
      tdm_load_1d_f32(&s_feat[fLo * 3],
                      &p[(size_t)b * (size_t)N * 3 + (size_t)(baseF + fLo) * 3],
                      (u32)((fHi - fLo) * 3));
    if (uHi > uLo)
      tdm_load_1d_f32(&s_unary[uLo],
                      &logits[(size_t)b * (size_t)N + (size_t)(base + uLo)],
                      (u32)(uHi - uLo));
    tdm_wait0();
  }
#else
  for (int i = t; i < FEATF; i += EXT) {
    int pos = baseF + i / 3;
    s_feat[i] = (pos >= 0 && pos < N)
                  ? p[(size_t)b * (size_t)N * 3 + (size_t)(baseF * 3 + i)]
                  : 0.0f;
  }
  {
    int g = base + t;
    s_unary[t] = (g >= 0 && g < N) ? logits[(size_t)b * (size_t)N + (size_t)g]
                                   : 0.0f;
  }
  if (t < RAD) {
    s_q[0][t] = 0.0f; s_q[0][EXT + RAD + t] = 0.0f;
    s_q[1][t] = 0.0f; s_q[1][EXT + RAD + t] = 0.0f;
  }
#endif
  __syncthreads();

  // ---- per-position Gaussian affinity weights (kept in registers) ----
  const int g = base + t;
  const int c = (t + RAD) * 3;
  const float fx = s_feat[c + 0];
  const float fy = s_feat[c + 1];
  const float fz = s_feat[c + 2];

  float w[2 * RAD];
  float wsum = 1e-8f;
#pragma unroll
  for (int j = 0; j < 2 * RAD; ++j) {
    const int dd = (j < RAD) ? (j - RAD) : (j - RAD + 1);  // -5..-1, 1..5
    const int m  = (t + RAD + dd) * 3;
    const float dx = fx - s_feat[m + 0];
    const float dy = fy - s_feat[m + 1];
    const float dz = fz - s_feat[m + 2];
    const int   gn = g + dd;
    const float mask = (gn >= 0 && gn < N) ? 1.0f : 0.0f;
    const float ww = __expf(-0.5f * (dx * dx + dy * dy + dz * dz)) * mask;
    w[j] = ww;
    wsum += ww;
  }
  const float inv_wsum = 1.0f / wsum;
  const float unary = s_unary[t];
  float q = 1.0f / (1.0f + __expf(-unary));   // q0 = sigmoid(unary)

  // ---- 5 fused mean-field iterations, double-buffered: 1 barrier/iter ----
  s_q[0][t + RAD] = q;
  __syncthreads();
#pragma unroll
  for (int it = 0; it < NITER; ++it) {
    const float* qr = s_q[it & 1];
    float msg = 0.0f;
#pragma unroll
    for (int j = 0; j < 2 * RAD; ++j) {
      const int dd = (j < RAD) ? (j - RAD) : (j - RAD + 1);
      msg += w[j] * qr[t + RAD + dd];
    }
    const float x = unary + msg * inv_wsum;
    q = 1.0f / (1.0f + __expf(-x));
    if (it + 1 < NITER) {
      s_q[(it + 1) & 1][t + RAD] = q;   // write opposite buffer: no WAR hazard
      __syncthreads();
    }
  }

  // ---- store valid interior of the trapezoid ----
  if (t >= NITER * RAD && t < NITER * RAD + TILE && g < N)
    out[(size_t)b * (size_t)N + (size_t)g] = q;
}

extern "C" void kernel_launch(void* const* d_in, const int* in_sizes, int n_in,
                              void* d_out, int out_size, void* d_ws, size_t ws_size,
                              hipStream_t stream) {
  (void)n_in; (void)d_ws; (void)ws_size; (void)out_size;
  const float* logits = (const float*)d_in[0];   // (16, N) f32
  const float* p      = (const float*)d_in[1];   // (16, N, 3) f32
  float*       out    = (float*)d_out;           // (16, N) f32

  const int B = 16;
  const int N = in_sizes[0] / B;                 // 100000
  const int nTiles = (N + TILE - 1) / TILE;

  dim3 grid((unsigned)nTiles, (unsigned)B, 1);
  crfrnn_fused_kernel<<<grid, EXT, 0, stream>>>(logits, p, out, N);
}